// SingleLayerGCN_52501680226427
// MI455X (gfx1250) — compile-verified
//
#include <hip/hip_runtime.h>
#include <hip/hip_bf16.h>

// ---------------------------------------------------------------------------
// GCNConv (512 -> 128) + log_softmax for MI455X (gfx1250, wave32).
//   h   = x @ W                      (WMMA f32 16x16x4)
//   out = b + h*dinv^2 (self loop)   (fused init)
//   out += scatter_add(h[src]*dinv[src]*dinv[dst] -> dst)   (f32 atomics, L2)
//   out = log_softmax(out, axis=1)   (wave-per-row, shfl_xor reductions)
// ---------------------------------------------------------------------------

typedef float v2f __attribute__((ext_vector_type(2)));
typedef float v8f __attribute__((ext_vector_type(8)));

#define N_NODES   50000
#define F_IN      512
#define F_OUT     128
#define M_TILES   (N_NODES / 16)   // 3125

// ---- degree / normalization ------------------------------------------------
__global__ void gcn_init_deg(float* __restrict__ deg, int n) {
  int i = blockIdx.x * blockDim.x + threadIdx.x;
  if (i < n) deg[i] = 1.0f;               // self-loop contributes 1
}

__global__ void gcn_accum_deg(const int* __restrict__ dst, float* __restrict__ deg, int e) {
  int i = blockIdx.x * blockDim.x + threadIdx.x;
  if (i < e) atomicAdd(&deg[dst[i]], 1.0f);
}

__global__ void gcn_dinv(float* __restrict__ deg, int n) {
  int i = blockIdx.x * blockDim.x + threadIdx.x;
  if (i < n) deg[i] = rsqrtf(deg[i]);     // deg >= 1 always (self-loops)
}

// ---- GEMM: h[N,128] = x[N,512] @ W[512,128] via V_WMMA_F32_16X16X4_F32 -----
// One wave computes a 16(M) x 128(N) slab: 8 accumulators of v8f.
// A-frag (16x4 f32, ISA 7.12.2): lanes 0-15 hold M=lane, K={k,k+1};
//                                lanes 16-31 hold M=lane-16, K={k+2,k+3}.
// B-frag (4x16 f32):             lanes 0-15 hold N=lane, K={k,k+1};
//                                lanes 16-31 hold N=lane-16, K={k+2,k+3}.
// D (16x16 f32): VGPR r -> M = r + 8*(lane>=16), N = lane&15.
__global__ __launch_bounds__(256) void gcn_gemm_wmma(const float* __restrict__ x,
                                                     const float* __restrict__ W,
                                                     float* __restrict__ h) {
  const int lane  = threadIdx.x & 31;
  const int wave  = threadIdx.x >> 5;
  const int mTile = blockIdx.x * 8 + wave;
  if (mTile >= M_TILES) return;           // wave-uniform: EXEC stays all-ones

  const int m0    = mTile * 16;
  const int rowA  = m0 + (lane & 15);
  const int kHalf = (lane >> 4) * 2;      // 0 or 2
  const int colB  = lane & 15;

  v8f acc[8];
#pragma unroll
  for (int t = 0; t < 8; ++t)
#pragma unroll
    for (int r = 0; r < 8; ++r) acc[t][r] = 0.0f;

  const float* __restrict__ xrow = x + (size_t)rowA * F_IN + kHalf;

  for (int k = 0; k < F_IN; k += 4) {
    const v2f a = *(const v2f*)(xrow + k);                 // global_load_b64
#pragma unroll
    for (int t = 0; t < 8; ++t) {
      const float* __restrict__ wp = W + (size_t)(k + kHalf) * F_OUT + t * 16 + colB;
      v2f bf;
      bf.x = wp[0];
      bf.y = wp[F_OUT];
      acc[t] = __builtin_amdgcn_wmma_f32_16x16x4_f32(
          /*neg_a=*/false, a, /*neg_b=*/false, bf,
          /*c_mod=*/(short)0, acc[t], /*reuse_a=*/false, /*reuse_b=*/false);
    }
  }

  const int rowD = m0 + (lane >> 4) * 8;
  const int colD = lane & 15;
#pragma unroll
  for (int t = 0; t < 8; ++t) {
    float* __restrict__ hp = h + (size_t)rowD * F_OUT + t * 16 + colD;
#pragma unroll
    for (int r = 0; r < 8; ++r) hp[(size_t)r * F_OUT] = acc[t][r];
  }
}

// ---- out = b + h * dinv^2  (self-loop term fused with bias) ----------------
__global__ void gcn_init_out(const float* __restrict__ h, const float* __restrict__ dinv,
                             const float* __restrict__ b, float* __restrict__ out, int total) {
  int i = blockIdx.x * blockDim.x + threadIdx.x;
  if (i >= total) return;
  int row = i >> 7, col = i & (F_OUT - 1);
  float di = dinv[row];
  out[i] = b[col] + h[i] * di * di;
}

// ---- edge scatter: one wave per edge, one float4 per lane ------------------
__global__ __launch_bounds__(256) void gcn_scatter(const int* __restrict__ src,
                                                   const int* __restrict__ dst,
                                                   const float* __restrict__ h,
                                                   const float* __restrict__ dinv,
                                                   float* __restrict__ out, int e) {
  const int edge = (int)((blockIdx.x * (size_t)blockDim.x + threadIdx.x) >> 5);
  const int lane = threadIdx.x & 31;
  if (edge >= e) return;
  const int   s   = src[edge];
  const int   d   = dst[edge];
  const float nrm = dinv[s] * dinv[d];
  const float4 v  = *(const float4*)(h + (size_t)s * F_OUT + lane * 4);  // b128 load
  float* __restrict__ o = out + (size_t)d * F_OUT + lane * 4;
  atomicAdd(o + 0, v.x * nrm);   // global_atomic_add_f32 (resolves in 192MB L2)
  atomicAdd(o + 1, v.y * nrm);
  atomicAdd(o + 2, v.z * nrm);
  atomicAdd(o + 3, v.w * nrm);
}

// ---- in-place log_softmax: one wave32 per row of 128 -----------------------
__global__ __launch_bounds__(256) void gcn_logsoftmax(float* __restrict__ out, int n) {
  const int row  = (int)((blockIdx.x * (size_t)blockDim.x + threadIdx.x) >> 5);
  const int lane = threadIdx.x & 31;
  if (row >= n) return;
  float* __restrict__ p = out + (size_t)row * F_OUT + lane * 4;
  float4 v = *(const float4*)p;

  float m = fmaxf(fmaxf(v.x, v.y), fmaxf(v.z, v.w));
#pragma unroll
  for (int off = 16; off > 0; off >>= 1) m = fmaxf(m, __shfl_xor(m, off, 32));

  float s = expf(v.x - m) + expf(v.y - m) + expf(v.z - m) + expf(v.w - m);
#pragma unroll
  for (int off = 16; off > 0; off >>= 1) s += __shfl_xor(s, off, 32);

  const float lse = m + logf(s);
  v.x -= lse; v.y -= lse; v.z -= lse; v.w -= lse;
  *(float4*)p = v;
}

// ---------------------------------------------------------------------------
extern "C" void kernel_launch(void* const* d_in, const int* in_sizes, int n_in,
                              void* d_out, int out_size, void* d_ws, size_t ws_size,
                              hipStream_t stream) {
  const float* x    = (const float*)d_in[0];   // [N, 512]
  const int*   eidx = (const int*)  d_in[1];   // [2, E]
  const float* W    = (const float*)d_in[2];   // [512, 128]
  const float* b    = (const float*)d_in[3];   // [128]
  float*       out  = (float*)d_out;           // [N, 128]

  const int N = in_sizes[0] / F_IN;
  const int E = in_sizes[1] / 2;
  const int* src = eidx;       // edge_index[0]
  const int* dst = eidx + E;   // edge_index[1]

  // workspace: dinv [N] then h [N,128], 256B aligned
  float* dinv = (float*)d_ws;
  size_t dinvBytes = (((size_t)N * sizeof(float)) + 255) & ~(size_t)255;
  float* h = (float*)((char*)d_ws + dinvBytes);

  // 1) degree -> dinv
  gcn_init_deg <<<(N + 255) / 256, 256, 0, stream>>>(dinv, N);
  gcn_accum_deg<<<(E + 255) / 256, 256, 0, stream>>>(dst, dinv, E);
  gcn_dinv     <<<(N + 255) / 256, 256, 0, stream>>>(dinv, N);

  // 2) h = x @ W  (WMMA; 8 waves/block, each wave owns 16x128)
  gcn_gemm_wmma<<<(M_TILES + 7) / 8, 256, 0, stream>>>(x, W, h);

  // 3) out = b + h*dinv^2
  const int total = N * F_OUT;
  gcn_init_out<<<(total + 255) / 256, 256, 0, stream>>>(h, dinv, b, out, total);

  // 4) edge scatter-add (wave per edge => 8 edges per 256-thread block)
  gcn_scatter<<<(E + 7) / 8, 256, 0, stream>>>(src, dst, h, dinv, out, E);

  // 5) log_softmax in place (wave per row)
  gcn_logsoftmax<<<(N + 7) / 8, 256, 0, stream>>>(out, N);
}